// LSTMSimple_51608327028781
// MI455X (gfx1250) — compile-verified
//
#include <hip/hip_runtime.h>
#include <hip/hip_bf16.h>

#define T_  512
#define B_  32
#define D_  512
#define H_  256
#define A1_ 16
#define A2_ 128
#define G4_ 1024  // 4*H

typedef __attribute__((ext_vector_type(2))) float v2f;
typedef __attribute__((ext_vector_type(4))) float v4f;
typedef __attribute__((ext_vector_type(8))) float v8f;

__device__ __forceinline__ v8f wmma_f32_16x16x4(v2f a, v2f b, v8f c) {
  // 8 args: (neg_a, A, neg_b, B, c_mod, C, reuse_a, reuse_b)
  return __builtin_amdgcn_wmma_f32_16x16x4_f32(false, a, false, b, (short)0, c,
                                               false, false);
}

// Branch-free tanh: native V_TANH_F32 on gfx1250 if the builtin exists,
// else 1 - 2/(exp(2x)+1) (one v_exp_f32, no EXEC-divergent slow path).
__device__ __forceinline__ float fast_tanh(float x) {
#if __has_builtin(__builtin_amdgcn_tanhf)
  return __builtin_amdgcn_tanhf(x);
#else
  float e = __expf(2.0f * x);
  return 1.0f - 2.0f / (e + 1.0f);
#endif
}

__device__ __forceinline__ float sigmoidf_(float x) {
  return 1.0f / (1.0f + __expf(-x));
}

// ---------------------------------------------------------------------------
// Phase 1: AttLayer1 over feature axis D. One wave per (t,b) row, single pass
// (reference uses exp with no max-subtract, so num/den accumulate directly).
// ---------------------------------------------------------------------------
__global__ __launch_bounds__(256) void att1_kernel(
    const float* __restrict__ x, const float* __restrict__ W1,
    const float* __restrict__ b1, const float* __restrict__ u1,
    float* __restrict__ seq) {
  const int lane = threadIdx.x & 31;
  const int wave = threadIdx.x >> 5;
  const int row  = blockIdx.x * 8 + wave;  // 0 .. T*B-1 (grid sized exactly)
  const float* xr = x + (size_t)row * D_;

  float w1v[A1_], b1v[A1_], u1v[A1_];
#pragma unroll
  for (int a = 0; a < A1_; ++a) { w1v[a] = W1[a]; b1v[a] = b1[a]; u1v[a] = u1[a]; }

  float num = 0.f, den = 0.f;
  for (int d = lane; d < D_; d += 32) {
    float xv = xr[d];
    float s = 0.f;
#pragma unroll
    for (int a = 0; a < A1_; ++a) s += fast_tanh(xv * w1v[a] + b1v[a]) * u1v[a];
    float e = __expf(s);
    den += e;
    num += xv * e;
  }
#pragma unroll
  for (int off = 16; off >= 1; off >>= 1) {
    num += __shfl_xor(num, off, 32);
    den += __shfl_xor(den, off, 32);
  }
  if (lane == 0) seq[row] = num / den;
}

// ---------------------------------------------------------------------------
// One-time pack of W_hh (4H x H, gate-major rows) into WMMA-B-fragment order:
//   PIF[(k4*2+kk)*H + j] = { Wt[k][i_j], Wt[k+1][i_j], Wt[k][f_j], Wt[k+1][f_j] }
//   PGO[(k4*2+kk)*H + j] = { same for gates g,o },   k = k4*4 + kk*2
// so one global_load_b128 yields two even-aligned v2f fragments with no movs.
// ---------------------------------------------------------------------------
__global__ __launch_bounds__(256) void pack_whh_kernel(
    const float* __restrict__ W_hh, float* __restrict__ P) {
  int idx = blockIdx.x * blockDim.x + threadIdx.x;  // 0 .. 4*H*H-1 (262144)
  int c    = idx & 3;           // element within quad
  int j    = (idx >> 2) & 255;  // hidden column
  int kk   = (idx >> 10) & 1;   // K half (lane>>4)
  int k4   = (idx >> 11) & 63;  // K quad
  int a    = idx >> 17;         // 0 = {i,f} array, 1 = {g,o} array
  int gate = a * 2 + (c >> 1);
  int k    = k4 * 4 + kk * 2 + (c & 1);
  P[idx] = W_hh[(gate * H_ + j) * H_ + k];
}

// One-time pack of W2 (H x A2) into v2f K-pair order:
//   W2P[2*((k4*2+kk)*A2 + a) + c] = W2[(k4*4+kk*2+c)*A2 + a]
__global__ __launch_bounds__(256) void pack_w2_kernel(
    const float* __restrict__ W2, float* __restrict__ W2P) {
  int idx = blockIdx.x * blockDim.x + threadIdx.x;  // 0 .. H*A2-1 (32768)
  int c  = idx & 1;
  int a  = (idx >> 1) & 127;
  int kk = (idx >> 8) & 1;
  int k4 = idx >> 9;
  W2P[idx] = W2[(k4 * 4 + kk * 2 + c) * A2_ + a];
}

// ---------------------------------------------------------------------------
// Phase 2: sequential LSTM, one workgroup = 32 waves on one WGP.
// Wave w owns batch-tile m=w&1 and j-columns [(w>>1)*16, +16) for ALL 4 gates
// (gate tiles are 16 n-tiles apart), so the nonlinearity runs on its own WMMA
// accumulators with c-state kept in registers. h lives in LDS.
// ---------------------------------------------------------------------------
__global__ __launch_bounds__(1024) void lstm_kernel(
    const float* __restrict__ seq, const float* __restrict__ W_ih,
    const float* __restrict__ W_hhP, const float* __restrict__ b_ih,
    const float* __restrict__ b_hh, const float* __restrict__ h0,
    const float* __restrict__ c0, float* __restrict__ hs) {
  __shared__ float h_s[B_][H_];     // 32 KB
  __shared__ float bias_s[G4_];     // 4 KB  (b_ih + b_hh)
  __shared__ float wih_s[G4_];      // 4 KB

  const int tid  = threadIdx.x;
  const int lane = tid & 31;
  const int wave = tid >> 5;   // 0..31
  const int m    = wave & 1;   // batch tile (rows m*16 .. m*16+15)
  const int nw   = wave >> 1;  // j-tile 0..15
  const int nc   = lane & 15;
  const int hi   = lane >> 4;  // half-wave: 0/1

  for (int i = tid; i < B_ * H_; i += 1024) h_s[i >> 8][i & 255] = h0[i];
  bias_s[tid] = b_ih[tid] + b_hh[tid];
  wih_s[tid]  = W_ih[tid];

  const int j = nw * 16 + nc;  // hidden column owned by this lane
  float c_reg[8];
#pragma unroll
  for (int g = 0; g < 8; ++g) {  // C/D layout: lanes<16 -> rows g, else g+8
    int b = m * 16 + hi * 8 + g;
    c_reg[g] = c0[b * H_ + j];
  }

  const int arow = m * 16 + nc;   // A-frag: row = lane%16 (both halves)
  const int kc0  = hi * 2;        // A/B frag: K = k4*4 + hi*2 + {0,1}
  const v4f* PIF = (const v4f*)W_hhP;          // {i,f} quads
  const v4f* PGO = PIF + (G4_ * H_ / 8);       // {g,o} quads
  const int bbase = hi * H_ + j;  // per-lane base within a k4 slab

  __syncthreads();

  for (int t = 0; t < T_; ++t) {
    v8f acc_i = {}; v8f acc_f = {}; v8f acc_g = {}; v8f acc_o = {};
#pragma unroll 4
    for (int k4 = 0; k4 < H_ / 4; ++k4) {
      int k = k4 * 4 + kc0;
      v2f a;
      a.x = h_s[arow][k];
      a.y = h_s[arow][k + 1];
      v4f qif = PIF[k4 * (2 * H_) + bbase];
      v4f qgo = PGO[k4 * (2 * H_) + bbase];
      v2f bi = __builtin_shufflevector(qif, qif, 0, 1);
      v2f bf = __builtin_shufflevector(qif, qif, 2, 3);
      v2f bg = __builtin_shufflevector(qgo, qgo, 0, 1);
      v2f bo = __builtin_shufflevector(qgo, qgo, 2, 3);
      acc_i = wmma_f32_16x16x4(a, bi, acc_i);
      acc_f = wmma_f32_16x16x4(a, bf, acc_f);
      acc_g = wmma_f32_16x16x4(a, bg, acc_g);
      acc_o = wmma_f32_16x16x4(a, bo, acc_o);
    }

    // x_t contribution: 8 consecutive batch entries per lane -> 2 vec4 loads
    const v4f* seqv = (const v4f*)(seq + t * B_);
    v4f s0 = seqv[m * 4 + hi * 2];
    v4f s1 = seqv[m * 4 + hi * 2 + 1];

    float h_new[8];
#pragma unroll
    for (int g = 0; g < 8; ++g) {
      int b = m * 16 + hi * 8 + g;
      float xt = (g < 4) ? s0[g] : s1[g - 4];
      float gi = acc_i[g] + xt * wih_s[j]          + bias_s[j];
      float gf = acc_f[g] + xt * wih_s[H_ + j]     + bias_s[H_ + j];
      float gg = acc_g[g] + xt * wih_s[2 * H_ + j] + bias_s[2 * H_ + j];
      float go = acc_o[g] + xt * wih_s[3 * H_ + j] + bias_s[3 * H_ + j];
      float cn = sigmoidf_(gf) * c_reg[g] + sigmoidf_(gi) * fast_tanh(gg);
      float hn = sigmoidf_(go) * fast_tanh(cn);
      c_reg[g] = cn;
      h_new[g] = hn;
      hs[(size_t)t * (B_ * H_) + b * H_ + j] = hn;
    }

    __syncthreads();  // all waves done reading old h from LDS
#pragma unroll
    for (int g = 0; g < 8; ++g) {
      int b = m * 16 + hi * 8 + g;
      h_s[b][j] = h_new[g];
    }
    __syncthreads();  // new h visible to every wave
  }
}

// ---------------------------------------------------------------------------
// Phase 3a: attention-2 scores. WMMA (16384x256)@(256x128); one wave per
// 16-row m-tile across all 8 n-tiles, then tanh*u2 row-reduce via shfl_xor.
// ---------------------------------------------------------------------------
__global__ __launch_bounds__(256) void att2_score_kernel(
    const float* __restrict__ hs, const float* __restrict__ W2P,
    const float* __restrict__ b2, const float* __restrict__ u2,
    float* __restrict__ e2) {
  const int lane  = threadIdx.x & 31;
  const int wave  = threadIdx.x >> 5;
  const int mtile = blockIdx.x * 8 + wave;  // 0..1023 (grid sized exactly)
  const int nc    = lane & 15;
  const int hi    = lane >> 4;
  const int kc0   = hi * 2;
  const float* arow = hs + (size_t)(mtile * 16 + nc) * H_;
  const v2f* P2 = (const v2f*)W2P;

  v8f acc[8];
#pragma unroll
  for (int n = 0; n < 8; ++n) acc[n] = v8f{};
  for (int k4 = 0; k4 < H_ / 4; ++k4) {
    int k = k4 * 4 + kc0;
    v2f a;
    a.x = arow[k];
    a.y = arow[k + 1];
    const v2f* prow = P2 + (k4 * 2 + hi) * A2_;
#pragma unroll
    for (int n = 0; n < 8; ++n) {
      v2f b = prow[n * 16 + nc];
      acc[n] = wmma_f32_16x16x4(a, b, acc[n]);
    }
  }
  float s[8];
#pragma unroll
  for (int g = 0; g < 8; ++g) s[g] = 0.f;
#pragma unroll
  for (int n = 0; n < 8; ++n) {
    int aidx = n * 16 + nc;
    float bb = b2[aidx], uu = u2[aidx];
#pragma unroll
    for (int g = 0; g < 8; ++g) s[g] += fast_tanh(acc[n][g] + bb) * uu;
  }
#pragma unroll
  for (int off = 1; off < 16; off <<= 1)
#pragma unroll
    for (int g = 0; g < 8; ++g) s[g] += __shfl_xor(s[g], off, 32);
  if (nc == 0) {
    int rbase = mtile * 16 + hi * 8;
#pragma unroll
    for (int g = 0; g < 8; ++g) e2[rbase + g] = __expf(s[g]);
  }
}

// Phase 3b: denominator over time, one wave per batch element.
__global__ __launch_bounds__(1024) void den_kernel(const float* __restrict__ e2,
                                                   float* __restrict__ den) {
  int lane = threadIdx.x & 31, b = threadIdx.x >> 5;
  float s = 0.f;
  for (int t = lane; t < T_; t += 32) s += e2[t * B_ + b];
#pragma unroll
  for (int off = 16; off >= 1; off >>= 1) s += __shfl_xor(s, off, 32);
  if (lane == 0) den[b] = s;
}

// Phase 3c: pooled[b][h] = sum_t hs[t][b][h] * e2[t][b] / den[b] (streams hs,
// 4 independent accumulators to keep loads in flight).
__global__ __launch_bounds__(256) void pool_kernel(
    const float* __restrict__ hs, const float* __restrict__ e2,
    const float* __restrict__ den, float* __restrict__ pooled) {
  int idx = blockIdx.x * blockDim.x + threadIdx.x;  // 0..B*H-1
  int b = idx >> 8;
  float a0 = 0.f, a1 = 0.f, a2 = 0.f, a3 = 0.f;
  for (int t = 0; t < T_; t += 4) {
    a0 += hs[(size_t)(t + 0) * (B_ * H_) + idx] * e2[(t + 0) * B_ + b];
    a1 += hs[(size_t)(t + 1) * (B_ * H_) + idx] * e2[(t + 1) * B_ + b];
    a2 += hs[(size_t)(t + 2) * (B_ * H_) + idx] * e2[(t + 2) * B_ + b];
    a3 += hs[(size_t)(t + 3) * (B_ * H_) + idx] * e2[(t + 3) * B_ + b];
  }
  pooled[idx] = ((a0 + a1) + (a2 + a3)) / den[b];
}

// Phase 3d: out[b] = pooled[b] . Wl + bl, one wave per batch element.
__global__ __launch_bounds__(1024) void head_kernel(
    const float* __restrict__ pooled, const float* __restrict__ Wl,
    const float* __restrict__ bl, float* __restrict__ out) {
  int lane = threadIdx.x & 31, b = threadIdx.x >> 5;
  float s = 0.f;
  for (int hh = lane; hh < H_; hh += 32) s += pooled[b * H_ + hh] * Wl[hh];
#pragma unroll
  for (int off = 16; off >= 1; off >>= 1) s += __shfl_xor(s, off, 32);
  if (lane == 0) out[b] = s + bl[0];
}

extern "C" void kernel_launch(void* const* d_in, const int* in_sizes, int n_in,
                              void* d_out, int out_size, void* d_ws,
                              size_t ws_size, hipStream_t stream) {
  const float* inputs = (const float*)d_in[0];
  const float* W1     = (const float*)d_in[1];
  const float* b1     = (const float*)d_in[2];
  const float* u1     = (const float*)d_in[3];
  const float* W_ih   = (const float*)d_in[4];
  const float* W_hh   = (const float*)d_in[5];
  const float* b_ih   = (const float*)d_in[6];
  const float* b_hh   = (const float*)d_in[7];
  const float* h0     = (const float*)d_in[8];
  const float* c0     = (const float*)d_in[9];
  const float* W2     = (const float*)d_in[10];
  const float* b2     = (const float*)d_in[11];
  const float* u2     = (const float*)d_in[12];
  const float* Wl     = (const float*)d_in[13];
  const float* bl     = (const float*)d_in[14];
  float* out = (float*)d_out;

  float* ws     = (float*)d_ws;
  float* seq    = ws;                  // T*B
  float* e2     = seq + T_ * B_;       // T*B
  float* den    = e2 + T_ * B_;        // B
  float* pooled = den + B_;            // B*H
  float* W_hhP  = pooled + B_ * H_;    // 4H*H packed quads ({i,f} then {g,o})
  float* W2P    = W_hhP + G4_ * H_;    // H*A2 packed K-pairs
  float* hs     = W2P + H_ * A2_;      // T*B*H  (~16 MB)

  att1_kernel<<<(T_ * B_) / 8, 256, 0, stream>>>(inputs, W1, b1, u1, seq);
  pack_whh_kernel<<<(G4_ * H_) / 256, 256, 0, stream>>>(W_hh, W_hhP);
  pack_w2_kernel<<<(H_ * A2_) / 256, 256, 0, stream>>>(W2, W2P);
  lstm_kernel<<<1, 1024, 0, stream>>>(seq, W_ih, W_hhP, b_ih, b_hh, h0, c0, hs);
  att2_score_kernel<<<(T_ * B_) / 128, 256, 0, stream>>>(hs, W2P, b2, u2, e2);
  den_kernel<<<1, 1024, 0, stream>>>(e2, den);
  pool_kernel<<<(B_ * H_) / 256, 256, 0, stream>>>(hs, e2, den, pooled);
  head_kernel<<<1, 1024, 0, stream>>>(pooled, Wl, bl, out);
}